// InstanceCriterion_71708773974465
// MI455X (gfx1250) — compile-verified
//
#include <hip/hip_runtime.h>
#include <hip/hip_bf16.h>
#include <math.h>

// Problem constants (from reference)
#define Bc   2
#define NQc  256
#define NTc  32
#define NPc  200000
#define NCc  19
#define EPSc 1.1920929e-07f
#define INFc 1.0e8f

// Tiling
#define KCHUNKS 50          // K chunks per (b, mtile)
#define CHUNK   4000        // NPc / KCHUNKS
#define PANEL   160         // K per LDS panel (4 waves x 40)
#define PANELS  25          // CHUNK / PANEL
#define LDS_STRIDE 164      // padded dword stride (mult of 4, conflict-free)

// Workspace layout (floats)
#define WS_SPOS 0
#define WS_SNEG 16384
#define WS_SSIG 32768
#define WS_NSUM 49152
#define WS_SSUM 49664
#define WS_TSUM 50176
#define WS_TOTAL 50240

typedef __attribute__((ext_vector_type(2))) float v2f;
typedef __attribute__((ext_vector_type(8))) float v8f;

#define LOG2E_F  1.44269504088896340736f
#define LN2_F    0.69314718055994530942f

#if __has_builtin(__builtin_amdgcn_exp2f)
#define FAST_EXP2(x) __builtin_amdgcn_exp2f(x)
#else
#define FAST_EXP2(x) exp2f(x)
#endif
#if __has_builtin(__builtin_amdgcn_logf)
#define FAST_LOG2(x) __builtin_amdgcn_logf(x)
#else
#define FAST_LOG2(x) log2f(x)
#endif
#if __has_builtin(__builtin_amdgcn_rcpf)
#define FAST_RCP(x) __builtin_amdgcn_rcpf(x)
#else
#define FAST_RCP(x) (1.0f / (x))
#endif

// Async global->LDS 16B copy (ASYNCcnt path) with synchronous fallback.
#if __has_builtin(__builtin_amdgcn_global_load_async_to_lds_b128)
#define HAVE_ASYNC_LDS 1
typedef int v4i_raw __attribute__((vector_size(16)));
typedef __attribute__((address_space(1))) v4i_raw gas_v4i;
typedef __attribute__((address_space(3))) v4i_raw las_v4i;
__device__ __forceinline__ void copy16_g2l(const void* src, void* dst) {
    __builtin_amdgcn_global_load_async_to_lds_b128((gas_v4i*)src, (las_v4i*)dst, 0, 0);
}
__device__ __forceinline__ void wait_async_keep15() { __builtin_amdgcn_s_wait_asynccnt(15); }
__device__ __forceinline__ void wait_async_all()    { __builtin_amdgcn_s_wait_asynccnt(0); }
#else
#define HAVE_ASYNC_LDS 0
__device__ __forceinline__ void copy16_g2l(const void* src, void* dst) {
    *reinterpret_cast<float4*>(dst) = *reinterpret_cast<const float4*>(src);
}
__device__ __forceinline__ void wait_async_keep15() {}
__device__ __forceinline__ void wait_async_all()    {}
#endif

// pos = softplus(-x), neg = softplus(x), sig = sigmoid(x). All via raw
// v_exp_f32 / v_log_f32 / v_rcp_f32 (1+t is in (1,2]: no denorm fixups needed).
__device__ __forceinline__ void transform3(float x, float& pos, float& neg, float& sig) {
    float ax  = fabsf(x);
    float t   = FAST_EXP2(ax * -LOG2E_F);      // exp(-|x|) in (0,1]
    float opt = 1.0f + t;                      // (1,2]
    float l1  = FAST_LOG2(opt) * LN2_F;        // ln(1+t) = softplus(-|x|)
    float mx  = fmaxf(x, 0.0f);
    neg = l1 + mx;                             // softplus(x)
    pos = neg - x;                             // softplus(-x)
    float r = FAST_RCP(opt);
    sig = (x >= 0.0f) ? r : t * r;             // sigmoid(x)
}

__global__ void __launch_bounds__(256) zero_ws_kernel(float* ws) {
    int i = blockIdx.x * 256 + threadIdx.x;
    if (i < WS_TOTAL) ws[i] = 0.0f;
}

__global__ void __launch_bounds__(256) tmsum_kernel(const int* __restrict__ tmask,
                                                    float* __restrict__ ws) {
    __shared__ int red[256];
    const int b = blockIdx.x >> 5;
    const int t = blockIdx.x & 31;
    const int* row = tmask + (size_t)(b * NTc + t) * NPc;
    int s = 0;
    for (int p = threadIdx.x; p < NPc; p += 256) s += row[p];
    red[threadIdx.x] = s;
    __syncthreads();
    for (int off = 128; off > 0; off >>= 1) {
        if (threadIdx.x < off) red[threadIdx.x] += red[threadIdx.x + off];
        __syncthreads();
    }
    if (threadIdx.x == 0) ws[WS_TSUM + b * NTc + t] = (float)red[0];
}

// One block = (batch, 16-row M tile, 4000-wide K chunk). 4 waves split K inside.
// Double-buffered LDS panels staged with GLOBAL_LOAD_ASYNC_TO_LDS_B128.
__global__ void __launch_bounds__(128) match_main_kernel(const float* __restrict__ qm,
                                                         const int*   __restrict__ tmask,
                                                         float* __restrict__ ws) {
    __shared__ float sA[2][16 * LDS_STRIDE];   // qm panel  16 x PANEL (f32)
    __shared__ int   sBi[2][32 * LDS_STRIDE];  // tm panel  32 x PANEL (raw int)

    const int bid   = blockIdx.x;
    const int b     = bid / (16 * KCHUNKS);
    const int mt    = (bid % (16 * KCHUNKS)) / KCHUNKS;
    const int kc    = bid % KCHUNKS;
    const int m0    = mt * 16;
    const int kbase = kc * CHUNK;

    const int tid  = threadIdx.x;
    const int lane = tid & 31;
    const int wave = tid >> 5;
    const int half = lane >> 4;      // 0: lanes 0-15, 1: lanes 16-31
    const int mrow = lane & 15;

    const float* qmB = qm    + (size_t)b * NQc * NPc + (size_t)m0 * NPc;
    const int*   tmB = tmask + (size_t)b * NTc * NPc;

    // 15 async b128 issues per wave per panel (5 for qm + 10 for tm)
    auto load_panel = [&](int p, int buf) {
        const int kp = kbase + p * PANEL;
        #pragma unroll
        for (int i = 0; i < 5; ++i) {              // 16 rows x 40 float4
            int idx = tid + i * 128;
            int r = idx / 40, c = idx % 40;
            copy16_g2l(qmB + (size_t)r * NPc + kp + c * 4, &sA[buf][r * LDS_STRIDE + c * 4]);
        }
        #pragma unroll
        for (int i = 0; i < 10; ++i) {             // 32 rows x 40 int4
            int idx = tid + i * 128;
            int r = idx / 40, c = idx % 40;
            copy16_g2l(tmB + (size_t)r * NPc + kp + c * 4,
                       &sBi[buf][r * LDS_STRIDE + c * 4]);
        }
    };

    v8f aP0 = {}, aP1 = {}, aN0 = {}, aN1 = {}, aS0 = {}, aS1 = {};
    float rneg = 0.0f, rsig = 0.0f;

    load_panel(0, 0);

    for (int p = 0; p < PANELS; ++p) {
        if (p + 1 < PANELS) {                      // prefetch next panel into other buffer
            load_panel(p + 1, (p + 1) & 1);
            wait_async_keep15();                   // panel p's 15 transfers complete
        } else {
            wait_async_all();
        }
        __syncthreads();                           // all waves' panel-p data visible

        const int buf = p & 1;
        const float* __restrict__ A  = sA[buf];
        const int*   __restrict__ Bi = sBi[buf];

        // ---- WMMA over this panel: wave w owns k-local [w*40, w*40+40) ----
        #pragma unroll
        for (int s = 0; s < 10; ++s) {
            const int kk = wave * 40 + s * 4 + 2 * half;   // this lane's K pair
            // A fragment (16x4 f32): lane holds qm[m0+mrow][kk], [kk+1]
            float x0 = A[mrow * LDS_STRIDE + kk];
            float x1 = A[mrow * LDS_STRIDE + kk + 1];
            float p0, n0, s0, p1, n1, s1;
            transform3(x0, p0, n0, s0);
            transform3(x1, p1, n1, s1);
            rneg += n0 + n1;
            rsig += s0 + s1;

            v2f ap; ap.x = p0; ap.y = p1;
            v2f an; an.x = n0; an.y = n1;
            v2f as; as.x = s0; as.y = s1;
            // B fragments (4x16 f32) for targets 0-15 and 16-31 (cvt int -> f32 on read)
            v2f b0, b1;
            b0.x = (float)Bi[mrow * LDS_STRIDE + kk];
            b0.y = (float)Bi[mrow * LDS_STRIDE + kk + 1];
            b1.x = (float)Bi[(mrow + 16) * LDS_STRIDE + kk];
            b1.y = (float)Bi[(mrow + 16) * LDS_STRIDE + kk + 1];

            aP0 = __builtin_amdgcn_wmma_f32_16x16x4_f32(false, ap, false, b0, (short)0, aP0, false, false);
            aP1 = __builtin_amdgcn_wmma_f32_16x16x4_f32(false, ap, false, b1, (short)0, aP1, false, false);
            aN0 = __builtin_amdgcn_wmma_f32_16x16x4_f32(false, an, false, b0, (short)0, aN0, false, false);
            aN1 = __builtin_amdgcn_wmma_f32_16x16x4_f32(false, an, false, b1, (short)0, aN1, false, false);
            aS0 = __builtin_amdgcn_wmma_f32_16x16x4_f32(false, as, false, b0, (short)0, aS0, false, false);
            aS1 = __builtin_amdgcn_wmma_f32_16x16x4_f32(false, as, false, b1, (short)0, aS1, false, false);
        }
        __syncthreads();                           // done reading buf before it is refilled
    }

    // ---- epilogue: C tile layout: value v of v8f sits at (M = v + 8*half, N = lane&15) ----
    float* Spos = ws + WS_SPOS;
    float* Sneg = ws + WS_SNEG;
    float* Ssig = ws + WS_SSIG;
    #pragma unroll
    for (int v = 0; v < 8; ++v) {
        int row  = m0 + v + 8 * half;
        int base = (b * NQc + row) * NTc + mrow;
        atomicAdd(&Spos[base],      aP0[v]);
        atomicAdd(&Spos[base + 16], aP1[v]);
        atomicAdd(&Sneg[base],      aN0[v]);
        atomicAdd(&Sneg[base + 16], aN1[v]);
        atomicAdd(&Ssig[base],      aS0[v]);
        atomicAdd(&Ssig[base + 16], aS1[v]);
    }
    atomicAdd(&ws[WS_NSUM + b * NQc + m0 + mrow], rneg);
    atomicAdd(&ws[WS_SSUM + b * NQc + m0 + mrow], rsig);
}

__device__ __forceinline__ float block_sum_256(float v, float* red) {
    int n = threadIdx.x;
    red[n] = v;
    __syncthreads();
    for (int off = 128; off > 0; off >>= 1) {
        if (n < off) red[n] += red[n + off];
        __syncthreads();
    }
    float r = red[0];
    __syncthreads();
    return r;
}

// One block per batch. Thread n owns query row n.
__global__ void __launch_bounds__(256) match_finalize_kernel(const float* __restrict__ ql,
                                                             const int*   __restrict__ tl,
                                                             const int*   __restrict__ tmask,
                                                             const int*   __restrict__ sel,
                                                             const float* __restrict__ ws,
                                                             float* __restrict__ out) {
    __shared__ float col[32 * 257];   // masked costs, column-major with pad
    __shared__ float red[256];
    __shared__ float thr[32];
    __shared__ int   ltl[32];
    __shared__ float stm[32];

    const int b = blockIdx.x;
    const int n = threadIdx.x;

    if (n < 32) {
        ltl[n] = tl[b * NTc + n];
        stm[n] = ws[WS_TSUM + b * NTc + n];
    }
    __syncthreads();

    // softmax stats for row n
    const float* qr = ql + (size_t)(b * NQc + n) * NCc;
    float mx = -3.0e38f;
    for (int c = 0; c < NCc; ++c) mx = fmaxf(mx, qr[c]);
    float se = 0.0f;
    for (int c = 0; c < NCc; ++c) se += FAST_EXP2((qr[c] - mx) * LOG2E_F);
    const float inv_se = FAST_RCP(se);

    const float ns = ws[WS_NSUM + b * NQc + n];
    const float ss = ws[WS_SSUM + b * NQc + n];
    const float* Pr = ws + WS_SPOS + (size_t)(b * NQc + n) * NTc;
    const float* Nr = ws + WS_SNEG + (size_t)(b * NQc + n) * NTc;
    const float* Sr = ws + WS_SSIG + (size_t)(b * NQc + n) * NTc;

    // pass 1: sums for means
    float sc = 0.0f, sb = 0.0f, sd = 0.0f;
    for (int t = 0; t < NTc; ++t) {
        float cls  = 1.0f - FAST_EXP2((qr[ltl[t]] - mx) * LOG2E_F) * inv_se;
        float bce  = (Pr[t] + (ns - Nr[t])) * (1.0f / (float)NPc);
        float dice = 1.0f - (2.0f * Sr[t] + 1.0f) * FAST_RCP(ss + stm[t] + 1.0f);
        sc += cls; sb += bce; sd += dice;
    }
    const float inv_cnt = 1.0f / (float)(NQc * NTc);
    float mc = block_sum_256(sc, red) * inv_cnt;
    float mb = block_sum_256(sb, red) * inv_cnt;
    float md = block_sum_256(sd, red) * inv_cnt;
    const float wc = 0.5f / (mc + EPSc);
    const float wb = 1.0f / (mb + EPSc);
    const float wd = 1.0f / (md + EPSc);

    // pass 2: final costs + masked column buffer
    const int sidx = sel[b * NQc + n];
    for (int t = 0; t < NTc; ++t) {
        float cls  = 1.0f - FAST_EXP2((qr[ltl[t]] - mx) * LOG2E_F) * inv_se;
        float bce  = (Pr[t] + (ns - Nr[t])) * (1.0f / (float)NPc);
        float dice = 1.0f - (2.0f * Sr[t] + 1.0f) * FAST_RCP(ss + stm[t] + 1.0f);
        float cost = wc * cls + wb * bce + wd * dice;
        out[(size_t)(b * NQc + n) * NTc + t] = cost;
        bool valid = tmask[(size_t)(b * NTc + t) * NPc + sidx] != 0;
        col[t * 257 + n] = valid ? cost : INFc;
    }
    __syncthreads();

    // per-target 2nd-smallest threshold over 256 queries
    if (n < 32) {
        float m1 = 3.0e38f, m2 = 3.0e38f;
        for (int q = 0; q < NQc; ++q) {
            float v = col[n * 257 + q];
            if (v < m1) { m2 = m1; m1 = v; }
            else if (v < m2) { m2 = v; }
        }
        thr[n] = m2;
    }
    __syncthreads();

    const size_t maskbase = (size_t)Bc * NQc * NTc;
    for (int t = 0; t < NTc; ++t) {
        float v = col[t * 257 + n];
        out[maskbase + (size_t)(b * NQc + n) * NTc + t] = (v < thr[t]) ? 1.0f : 0.0f;
    }
}

extern "C" void kernel_launch(void* const* d_in, const int* in_sizes, int n_in,
                              void* d_out, int out_size, void* d_ws, size_t ws_size,
                              hipStream_t stream) {
    (void)in_sizes; (void)n_in; (void)out_size; (void)ws_size;
    const float* ql  = (const float*)d_in[0];   // (B, NQ, NC) f32
    const float* qm  = (const float*)d_in[1];   // (B, NQ, NP) f32
    const int*   tl  = (const int*)d_in[2];     // (B, NT) int
    const int*   tmk = (const int*)d_in[3];     // (B, NT, NP) bool->int
    const int*   sel = (const int*)d_in[4];     // (B, NQ) int
    float* out = (float*)d_out;
    float* ws  = (float*)d_ws;

    zero_ws_kernel<<<(WS_TOTAL + 255) / 256, 256, 0, stream>>>(ws);
    tmsum_kernel<<<Bc * NTc, 256, 0, stream>>>(tmk, ws);
    match_main_kernel<<<Bc * 16 * KCHUNKS, 128, 0, stream>>>(qm, tmk, ws);
    match_finalize_kernel<<<Bc, 256, 0, stream>>>(ql, tl, tmk, sel, ws, out);
}